// MPSLayer_66692252172373
// MI455X (gfx1250) — compile-verified
//
#include <hip/hip_runtime.h>

typedef __attribute__((ext_vector_type(2))) float v2f;
typedef __attribute__((ext_vector_type(4))) float v4f;
typedef __attribute__((ext_vector_type(8))) float v8f;

#define BATCH 512
#define SITES 64
#define FDIM  16
#define DDIM  64
#define BT    16                      // batch rows per workgroup
#define KTOT  (DDIM * FDIM)           // 1024 contracted (d,f) pairs per site
#define SITE_STRIDE (DDIM * FDIM * DDIM)  // 65536 floats per tmid site

// D = vx(16x1024) @ tmid_s(1024x64), site-chained; fp32 WMMA 16x16x4.
__global__ __launch_bounds__(256, 1) void mps_chain_kernel(
    const float* __restrict__ x,      // [B,S,F]
    const float* __restrict__ t0,     // [F,D]
    const float* __restrict__ tmid,   // [S-2,D,F,D]
    const float* __restrict__ tlast,  // [D,F]
    float* __restrict__ out)          // [B]
{
    __shared__ float vld[BT * DDIM];   // current v, [m][e]
    __shared__ float cbuf[BT * DDIM];  // K-split partial C / final reduce scratch

    const int tid  = threadIdx.x;
    const int lane = tid & 31;
    const int wave = tid >> 5;         // 8 waves
    const int w_e  = wave & 3;         // e-tile 0..3
    const int w_k  = wave >> 2;        // K half 0..1
    const int lm   = lane & 15;        // row (A/M, or col N) within tile
    const int lh   = lane >> 4;        // lane half selects K sub-rows
    const int b0   = blockIdx.x * BT;

    const int m  = tid >> 4;           // helper-phase row 0..15
    const int tr = tid & 15;           // helper-phase sub-index

    // ---------------- init: v[m][e] = sum_f x[b0+m,0,f] * t0[f][e] ----------------
    {
        const float* xrow = x + (size_t)(b0 + m) * (SITES * FDIM);
        const int e0 = tr * 4;
        v4f acc = {};
        #pragma unroll
        for (int f = 0; f < FDIM; ++f)
            acc += xrow[f] * *(const v4f*)(t0 + f * DDIM + e0);
        *(v4f*)(vld + m * DDIM + e0) = acc;
    }
    __syncthreads();

    const int col = w_e * 16 + lm;     // output column e for this lane

    // ---------------- site chain ----------------
    for (int s = 0; s < SITES - 2; ++s) {
        // prefetch next site's 256KB tmid slice into L2 (4096 lines / 256 threads)
        if (s + 1 < SITES - 2) {
            const char* pn = (const char*)(tmid + (size_t)(s + 1) * SITE_STRIDE);
            #pragma unroll
            for (int i = 0; i < 16; ++i)
                __builtin_prefetch(pn + ((size_t)(tid + i * 256)) * 64, 0, 1);
        }

        // per-lane x slice for this site, kept in registers (A-fragment factor)
        const float* xrow = x + (size_t)(b0 + lm) * (SITES * FDIM) + (size_t)(s + 1) * FDIM;
        v4f xv0 = *(const v4f*)(xrow + 0);
        v4f xv1 = *(const v4f*)(xrow + 4);
        v4f xv2 = *(const v4f*)(xrow + 8);
        v4f xv3 = *(const v4f*)(xrow + 12);

        // B stream: tmid_s rows (w_k*512 + 2*lh + 4k), column col
        const float* bp = tmid + (size_t)s * SITE_STRIDE
                        + (size_t)(w_k * 512 + 2 * lh) * DDIM + col;
        const float* vrow = vld + lm * DDIM + w_k * 32;

        v8f acc = {};
        #pragma unroll 2
        for (int kb = 0; kb < 32; ++kb) {          // 4 K-steps per kb -> 128 WMMAs
            const float vd = vrow[kb];             // v[lm, d], d = w_k*32 + kb
            #pragma unroll
            for (int i = 0; i < 4; ++i) {
                v4f xq = (i == 0) ? xv0 : (i == 1) ? xv1 : (i == 2) ? xv2 : xv3;
                float x0 = lh ? xq.z : xq.x;       // f = 4i + 2*lh
                float x1 = lh ? xq.w : xq.y;       // f = 4i + 2*lh + 1
                v2f a; a.x = vd * x0; a.y = vd * x1;   // vx outer product on the fly
                v2f b; b.x = bp[0];  b.y = bp[DDIM];
                bp += 4 * DDIM;
                acc = __builtin_amdgcn_wmma_f32_16x16x4_f32(
                        false, a, false, b, (short)0, acc, false, false);
            }
        }

        // reduce the two K-half partials; C layout: vgpr j -> M = j + 8*lh, N = lm
        if (w_k == 0) {
            #pragma unroll
            for (int j = 0; j < 8; ++j)
                cbuf[(lh * 8 + j) * DDIM + col] = acc[j];
        }
        __syncthreads();
        if (w_k == 1) {
            #pragma unroll
            for (int j = 0; j < 8; ++j)
                vld[(lh * 8 + j) * DDIM + col] = cbuf[(lh * 8 + j) * DDIM + col] + acc[j];
        }
        __syncthreads();
    }

    // ---------------- final: out[b] = sum_e v[b,e] * (sum_f x[b,S-1,f]*tlast[e,f]) ----
    {
        const float* xrow = x + (size_t)(b0 + m) * (SITES * FDIM) + (size_t)(SITES - 1) * FDIM;
        v4f xl0 = *(const v4f*)(xrow + 0);
        v4f xl1 = *(const v4f*)(xrow + 4);
        v4f xl2 = *(const v4f*)(xrow + 8);
        v4f xl3 = *(const v4f*)(xrow + 12);
        float partial = 0.f;
        #pragma unroll
        for (int j = 0; j < 4; ++j) {
            const int e = tr * 4 + j;
            const float* tl = tlast + e * FDIM;
            v4f p = xl0 * *(const v4f*)(tl + 0) + xl1 * *(const v4f*)(tl + 4)
                  + xl2 * *(const v4f*)(tl + 8) + xl3 * *(const v4f*)(tl + 12);
            partial += vld[m * DDIM + e] * (p.x + p.y + p.z + p.w);
        }
        cbuf[m * 16 + tr] = partial;
    }
    __syncthreads();
    if (tid < BT) {
        float ssum = 0.f;
        #pragma unroll
        for (int i = 0; i < 16; ++i) ssum += cbuf[tid * 16 + i];
        out[b0 + tid] = ssum;
    }
}

extern "C" void kernel_launch(void* const* d_in, const int* in_sizes, int n_in,
                              void* d_out, int out_size, void* d_ws, size_t ws_size,
                              hipStream_t stream) {
    const float* x     = (const float*)d_in[0];  // [512,64,16]
    const float* t0    = (const float*)d_in[1];  // [16,64]
    const float* tmid  = (const float*)d_in[2];  // [62,64,16,64]
    const float* tlast = (const float*)d_in[3];  // [64,16]
    float* out = (float*)d_out;                  // [512]
    mps_chain_kernel<<<dim3(BATCH / BT), dim3(256), 0, stream>>>(x, t0, tmid, tlast, out);
}